// MVLiftingModule_77653008711906
// MI455X (gfx1250) — compile-verified
//
#include <hip/hip_runtime.h>
#include <cstdint>

// Problem constants (from reference setup_inputs)
#define BB 4
#define NN 2048
#define VV 12
#define CC 50
#define PP 16384      // H*W
#define TILE 256      // pixels per block in the streaming kernel
#define TILES_PER_BV (PP / TILE)   // 64

typedef __attribute__((ext_vector_type(4))) uint32_t u32x4;
typedef __attribute__((ext_vector_type(8))) int32_t  i32x8;
typedef __attribute__((ext_vector_type(4))) int32_t  i32x4;

#if defined(__has_builtin)
# if __has_builtin(__builtin_amdgcn_tensor_load_to_lds) && __has_builtin(__builtin_amdgcn_s_wait_tensorcnt)
#  define HAS_TDM 1
# endif
#endif

// ---------------------------------------------------------------------------
// Kernel 2 (placed first so the disasm snippet shows the TDM emission):
// stream logits (157 MB) tile-by-tile via the Tensor Data Mover, per-pixel
// argmax over C, conditional atomicMax scatter of key = am*P + p.
// One block = one (b,v) slab x 256-pixel tile; TDM moves the 50x256 f32 tile
// (row stride P) into LDS with a single 2D descriptor.
// ---------------------------------------------------------------------------
__global__ __launch_bounds__(TILE) void mvlift_argmax_scatter(
    const float* __restrict__ pred,       // (B,V,C,P)
    const int*   __restrict__ p2p,        // (B,V,P)
    const int*   __restrict__ parts_nb,   // (B,)
    int*         __restrict__ winner) {   // (B,V,N)
  __shared__ float tile[CC * TILE];

  const int bx      = blockIdx.x;
  const int tileIdx = bx & (TILES_PER_BV - 1);
  const int bv      = bx >> 6;            // log2(TILES_PER_BV) = 6
  const int b       = bv / VV;
  const int p0      = tileIdx * TILE;
  const int t       = threadIdx.x;

  // Independent loads issued early: in flight (LOADcnt/KMcnt) while the TDM
  // transfer (TENSORcnt) completes.
  const int pp    = p2p[bv * PP + p0 + t];
  const int maxcl = parts_nb[b];

#ifdef HAS_TDM
  if (t < 32) {  // wave 0 only: TDM ignores EXEC, each wave would re-issue
    const uint64_t gaddr = (uint64_t)(uintptr_t)(pred + (size_t)bv * CC * PP + p0);
    const uint32_t ldsa  = (uint32_t)(uintptr_t)(&tile[0]); // low 32 bits = LDS offset

    u32x4 g0;
    g0.x = 1u;                                              // count=1, user desc
    g0.y = ldsa;                                            // lds_addr (bytes)
    g0.z = (uint32_t)gaddr;                                 // global_addr[31:0]
    g0.w = ((uint32_t)(gaddr >> 32) & 0x01FFFFFFu)          // global_addr[56:32]
         | (2u << 30);                                      // type = 2 ("image")

    i32x8 g1;
    g1[0] = (int)(2u << 16);                                // data_size = 4 bytes
    g1[1] = (int)(((uint32_t)PP & 0xFFFFu) << 16);          // tensor_dim0[15:0]
    g1[2] = (int)(((uint32_t)PP >> 16) | ((uint32_t)CC << 16)); // dim0[31:16] | dim1[15:0]
    g1[3] = (int)((uint32_t)TILE << 16);                    // dim1[31:16]=0 | tile_dim0
    g1[4] = (int)(uint32_t)CC;                              // tile_dim1 | tile_dim2=0
    g1[5] = (int)(uint32_t)PP;                              // tensor_dim0_stride[31:0]
    g1[6] = 0;                                              // stride hi | dim1_stride lo
    g1[7] = 0;

    i32x4 g2 = {0, 0, 0, 0};
    i32x4 g3 = {0, 0, 0, 0};
    i32x8 g4 = {0, 0, 0, 0, 0, 0, 0, 0};   // 6-arg toolchain: extra group, zero-filled
    __builtin_amdgcn_tensor_load_to_lds(g0, g1, g2, g3, g4, 0);
    __builtin_amdgcn_s_wait_tensorcnt((short)0);
  }
  __syncthreads();
#else
  // Fallback: coalesced global->LDS fill
  {
    const float* base = pred + (size_t)bv * CC * PP + p0;
    for (int i = t; i < CC * TILE; i += TILE) {
      int c = i / TILE, x = i - c * TILE;
      tile[i] = base[(size_t)c * PP + x];
    }
  }
  __syncthreads();
#endif

  float m  = tile[t];
  int   am = 0;
#pragma unroll
  for (int c = 1; c < CC; ++c) {
    float x = tile[c * TILE + t];
    if (x > m) { m = x; am = c; }   // strict '>' => first-occurrence argmax
  }
  if (pp >= 0 && am >= 1 && am <= maxcl) {
    atomicMax(&winner[bv * NN + pp], am * PP + (p0 + t));
  }
}

// ---------------------------------------------------------------------------
// Kernel 1: winner[b,v,n] = -1
// ---------------------------------------------------------------------------
__global__ void mvlift_init_winner(int* __restrict__ winner, int n) {
  int i = blockIdx.x * blockDim.x + threadIdx.x;
  if (i < n) winner[i] = -1;
}

// ---------------------------------------------------------------------------
// Kernel 3: per point (b,n): for each view with winner>=0, gather the 50
// logits at the winning pixel, softmax, weight by views_weights, accumulate;
// normalize by max(#contributing views, 1). Gathers mostly hit L2 (192 MB
// holds the 157 MB logits just streamed).
// ---------------------------------------------------------------------------
__global__ void mvlift_gather_out(
    const float* __restrict__ pred,      // (B,V,C,P)
    const float* __restrict__ vw,        // (B,V)
    const int*   __restrict__ winner,    // (B,V,N)
    float*       __restrict__ out) {     // (B,C,N)
  int idx = blockIdx.x * blockDim.x + threadIdx.x;
  if (idx >= BB * NN) return;
  int b = idx / NN;
  int n = idx - b * NN;

  float acc[CC];
#pragma unroll
  for (int c = 0; c < CC; ++c) acc[c] = 0.0f;
  int cnt = 0;

  for (int v = 0; v < VV; ++v) {
    int w = winner[(b * VV + v) * NN + n];
    if (w < 0) continue;
    ++cnt;
    int p = w & (PP - 1);                // winner % P (winner >= 0, P pow2)
    const float* base = pred + ((size_t)(b * VV + v) * CC) * PP + p;

    float x[CC];
#pragma unroll
    for (int c = 0; c < CC; ++c) x[c] = base[(size_t)c * PP];

    float m = x[0];
#pragma unroll
    for (int c = 1; c < CC; ++c) m = fmaxf(m, x[c]);

    float s = 0.0f;
#pragma unroll
    for (int c = 0; c < CC; ++c) { float e = __expf(x[c] - m); x[c] = e; s += e; }

    float r = vw[b * VV + v] / s;
#pragma unroll
    for (int c = 0; c < CC; ++c) acc[c] += x[c] * r;
  }

  float inv = 1.0f / (float)(cnt > 0 ? cnt : 1);
#pragma unroll
  for (int c = 0; c < CC; ++c)
    out[((size_t)b * CC + c) * NN + n] = acc[c] * inv;
}

// ---------------------------------------------------------------------------
extern "C" void kernel_launch(void* const* d_in, const int* in_sizes, int n_in,
                              void* d_out, int out_size, void* d_ws, size_t ws_size,
                              hipStream_t stream) {
  // inputs (setup_inputs order): points(unused), predictions_2d,
  // rendered_pix_to_point, views_weights, parts_nb
  const float* pred     = (const float*)d_in[1];
  const int*   p2p      = (const int*)  d_in[2];
  const float* vw       = (const float*)d_in[3];
  const int*   parts_nb = (const int*)  d_in[4];
  float*       out      = (float*)d_out;
  int*         winner   = (int*)d_ws;          // B*V*N ints = 384 KiB

  const int wn = BB * VV * NN;
  mvlift_init_winner<<<(wn + 255) / 256, 256, 0, stream>>>(winner, wn);

  mvlift_argmax_scatter<<<BB * VV * TILES_PER_BV, TILE, 0, stream>>>(
      pred, p2p, parts_nb, winner);

  mvlift_gather_out<<<(BB * NN + 255) / 256, 256, 0, stream>>>(
      pred, vw, winner, out);
}